// TemporalWeightedMeanAggregator_23510650978750
// MI455X (gfx1250) — compile-verified
//
#include <hip/hip_runtime.h>
#include <stdint.h>

// ---------------------------------------------------------------------------
// TemporalWeightedMeanAggregator for MI455X (gfx1250)
//   N = 1,048,576 messages, D = 256, S = 65,536 segments (derived at launch)
//   HBM floor ~46us (1 GiB message stream); wsum/tw stay resident in 192MB L2.
//   CDNA5 paths:
//     * global_load_async_to_lds_b128 double-buffered tile staging (ASYNCcnt)
//     * V_WMMA_F32_16X16X4_F32 chain applies the diag(w) pre-aggregation
//       matrix per 16-row tile (scaling runs on the tensor pipe)
//     * clause'd f32 global atomics, DEV scope (L2-resident scatter-reduce)
//   This rev: no storecnt drain per tile (wave-level LDS ordering instead of
//   __syncthreads fence), fully unrolled column loop so all DS/atomic column
//   offsets fold into immediate offset fields.
// ---------------------------------------------------------------------------

#define BETA 0.8f
#define D_MSG 256
#define TILE_ROWS 16
#define TILES_PER_BLOCK 16   // rows per block = 256

typedef float v2f __attribute__((ext_vector_type(2)));
typedef float v8f __attribute__((ext_vector_type(8)));

__device__ __forceinline__ void gatomic_add(float* p, float v) {
  __hip_atomic_fetch_add(p, v, __ATOMIC_RELAXED, __HIP_MEMORY_SCOPE_AGENT);
}

// Compiler-only ordering for same-wave LDS store->load (LDS unit is in-order
// per wave; no s_wait / no storecnt drain needed, unlike __syncthreads()).
__device__ __forceinline__ void wave_lds_order() {
  asm volatile("" ::: "memory");
  __builtin_amdgcn_wave_barrier();
  asm volatile("" ::: "memory");
}

// Stage one 16x256 f32 tile (16 KB) into LDS with async loads.
// 32 instructions x (32 lanes x 16B) = 16 KB; tracked on ASYNCcnt.
__device__ __forceinline__ void stage_async(const float* gsrc, const float* lbuf,
                                            int lane) {
  unsigned long long ga =
      (unsigned long long)(uintptr_t)gsrc + (unsigned)lane * 16u;
  unsigned la = (unsigned)(uintptr_t)lbuf + (unsigned)lane * 16u; // LDS off = low 32b of flat
#pragma unroll
  for (int j = 0; j < 32; ++j) {
    asm volatile("global_load_async_to_lds_b128 %0, %1, off offset:%2"
                 :: "v"(la), "v"(ga), "i"(j * 512) : "memory");
  }
}

// ---------------------------------------------------------------------------
__global__ void k_zero_f4(float4* __restrict__ p, long n4) {
  long i = (long)blockIdx.x * blockDim.x + threadIdx.x;
  long stride = (long)gridDim.x * blockDim.x;
  for (; i < n4; i += stride) p[i] = make_float4(0.f, 0.f, 0.f, 0.f);
}

// segment max of timestamps via uint-bit atomicMax (valid: timestamps >= 0)
__global__ void k_seg_max(const float* __restrict__ ts, const int* __restrict__ ids,
                          unsigned* __restrict__ last_bits, int n) {
  int i = blockIdx.x * blockDim.x + threadIdx.x;
  if (i < n) atomicMax(&last_bits[ids[i]], __float_as_uint(ts[i]));
}

// ---------------------------------------------------------------------------
// Main scatter-reduce. One wave per block; 16 tiles of 16 rows each.
__global__ __launch_bounds__(32)
void k_accum(const float* __restrict__ msgs, const float* __restrict__ ts,
             const int* __restrict__ ids, const unsigned* __restrict__ last_bits,
             float* __restrict__ agg, float* __restrict__ tw) {
  __shared__ float stage[2][TILE_ROWS * D_MSG];   // 32 KB double buffer
  __shared__ float w_lds[TILE_ROWS];
  __shared__ unsigned off_lds[TILE_ROWS];

  const int lane = threadIdx.x;      // 0..31
  const int l15  = lane & 15;
  const int hi   = lane >> 4;        // 0: lanes 0-15, 1: lanes 16-31

  const long block_row0 = (long)blockIdx.x * (TILES_PER_BLOCK * TILE_ROWS);

  // Prologue: stage tile 0 into buffer 0.
  stage_async(msgs + block_row0 * D_MSG, stage[0], lane);

#pragma unroll 1
  for (int t = 0; t < TILES_PER_BLOCK; ++t) {
    const long row0 = block_row0 + (long)t * TILE_ROWS;
    const int buf = t & 1;

    // Issue next tile's async loads before touching this tile (latency hiding;
    // ASYNCcnt completes in order, so waiting to <=32 later implies this
    // tile's 32 loads have landed).
    if (t + 1 < TILES_PER_BLOCK)
      stage_async(msgs + (row0 + TILE_ROWS) * D_MSG, stage[buf ^ 1], lane);

    // Per-row weights + tw scatter (lanes 0-15; one row each).
    if (lane < 16) {
      float tv = ts[row0 + lane];
      int id = ids[row0 + lane];
      float last = __uint_as_float(last_bits[id]);
      float w = __expf(BETA * (tv - last));   // <= 1, underflows to 0 for old msgs
      w_lds[lane] = w;
      off_lds[lane] = (unsigned)id * D_MSG;
      gatomic_add(&tw[id], w);
    }
    wave_lds_order();   // same-wave LDS in-order; no fence, no storecnt drain

    // A fragments for P = diag(w), split into 4 one-hot K-blocks (K=4 WMMA).
    // 16x4 f32 A layout: lanes0-15 row=l15 K={0(.x),1(.y)}, lanes16-31 K={2,3}.
    const float wv = w_lds[l15];
    v2f a[4];
#pragma unroll
    for (int b = 0; b < 4; ++b) {
      a[b].x = (l15 == 4 * b + 2 * hi)     ? wv : 0.f;
      a[b].y = (l15 == 4 * b + 2 * hi + 1) ? wv : 0.f;
    }
    // Tile-invariant atomic base pointers (C row M = v + 8*hi).
    float* bases[8];
#pragma unroll
    for (int v = 0; v < 8; ++v)
      bases[v] = agg + (size_t)off_lds[v + 8 * hi] + l15;

    // Wait until the *current* tile's 32 async loads have landed in LDS.
    if (t + 1 < TILES_PER_BLOCK)
      asm volatile("s_wait_asynccnt 32" ::: "memory");
    else
      asm volatile("s_wait_asynccnt 0" ::: "memory");

    const float* sb = stage[buf] + (2 * hi) * D_MSG + l15;  // per-lane DS base

    // 16 column tiles of 16: C(16x16) = diag(w) * M_tile via 4 chained WMMAs,
    // then scatter C with f32 atomics (L2 resident). Fully unrolled so the
    // column offsets fold into DS / global-atomic immediate offset fields.
#pragma unroll
    for (int ct = 0; ct < 16; ++ct) {
      const int cb = ct * 16;
      v8f c = {};
#pragma unroll
      for (int b = 0; b < 4; ++b) {
        // 4x16 f32 B layout mirrors A's K split: .x K={0|2}, .y K={1|3}
        v2f bb;
        bb.x = sb[4 * b * D_MSG + cb];
        bb.y = sb[(4 * b + 1) * D_MSG + cb];
        c = __builtin_amdgcn_wmma_f32_16x16x4_f32(false, a[b], false, bb,
                                                  (short)0, c, false, false);
      }
#pragma unroll
      for (int v = 0; v < 8; ++v)
        gatomic_add(bases[v] + cb, c[v]);
    }
    wave_lds_order();   // protect w_lds/off_lds before next tile overwrites
  }
}

// ---------------------------------------------------------------------------
// agg[s,:] *= 1/max(tw,1e-30) (rows of empty segments are already zero);
// last_t_out[s] = tw>0 ? last_t[s] : 0.  One wave per segment row.
__global__ void k_finalize(float* __restrict__ agg, const float* __restrict__ tw,
                           const unsigned* __restrict__ last_bits,
                           float* __restrict__ last_out, int S) {
  int w = (blockIdx.x * blockDim.x + threadIdx.x) >> 5;
  int lane = threadIdx.x & 31;
  if (w >= S) return;
  float t = tw[w];
  bool nz = t > 0.f;
  float inv = nz ? 1.0f / fmaxf(t, 1e-30f) : 0.f;   // empty rows: 0*0 = 0
  float4* row = (float4*)(agg + (long)w * D_MSG) + lane * 2;
  float4 x0 = row[0], x1 = row[1];
  x0.x *= inv; x0.y *= inv; x0.z *= inv; x0.w *= inv;
  x1.x *= inv; x1.y *= inv; x1.z *= inv; x1.w *= inv;
  row[0] = x0; row[1] = x1;
  if (lane == 0) last_out[w] = nz ? __uint_as_float(last_bits[w]) : 0.f;
}

// ---------------------------------------------------------------------------
extern "C" void kernel_launch(void* const* d_in, const int* in_sizes, int n_in,
                              void* d_out, int out_size, void* d_ws, size_t ws_size,
                              hipStream_t stream) {
  const float* msgs = (const float*)d_in[0];
  const float* ts   = (const float*)d_in[1];
  const int*   ids  = (const int*)d_in[2];

  const int N = in_sizes[1];                 // 1,048,576
  const int D = in_sizes[0] / N;             // 256 (tiling assumes this)
  const int S = out_size / (D + 1);          // 65,536

  float* agg = (float*)d_out;                // [S*D] accumulate in place
  float* last_out = agg + (long)S * D;       // [S]
  unsigned* last_bits = (unsigned*)d_ws;     // [S]  (bits of f32, init 0 == 0.0f)
  float* twp = (float*)d_ws + S;             // [S]

  // Re-init accumulators every call (harness does not re-poison between replays).
  k_zero_f4<<<8192, 256, 0, stream>>>((float4*)agg, (long)S * D / 4);
  k_zero_f4<<<256, 256, 0, stream>>>((float4*)d_ws, (long)2 * S / 4);

  k_seg_max<<<(N + 255) / 256, 256, 0, stream>>>(ts, ids, last_bits, N);

  const int rows_per_block = TILES_PER_BLOCK * TILE_ROWS;     // 256
  k_accum<<<N / rows_per_block, 32, 0, stream>>>(msgs, ts, ids, last_bits,
                                                 agg, twp);

  k_finalize<<<(S * 32 + 255) / 256, 256, 0, stream>>>(agg, twp, last_bits,
                                                       last_out, S);
}